// GCBlock_81003083203735
// MI455X (gfx1250) — compile-verified
//
#include <hip/hip_runtime.h>
#include <math.h>

// Problem constants (from reference setup_inputs)
#define BB   4
#define HH   256
#define WW   256
#define CC   256
#define MID  64
#define HWN  (HH*WW)          // 65536 spatial positions per batch
#define NPOS (BB*HWN)         // 262144 total positions
#define CHUNKS 256            // pooling chunks per batch (HWN/256)

typedef __attribute__((ext_vector_type(2))) float v2f;
typedef __attribute__((ext_vector_type(8))) float v8f;

// ---------------------------------------------------------------------------
// K1: logits[n] = x[n,:].Wv + pe[n % HW] * sum(Wv)
// 1 thread per position, float4 row loads, Wv staged in LDS.
// ---------------------------------------------------------------------------
__global__ void k_logits(const float* __restrict__ x,
                         const float* __restrict__ pe,
                         const float* __restrict__ Wv,
                         float* __restrict__ logits) {
    __shared__ float wv_s[CC];
    __shared__ float red[256];
    const int tid = threadIdx.x;
    wv_s[tid] = Wv[tid];
    red[tid]  = wv_s[tid];
    __syncthreads();
    // tree-reduce sum(Wv)
    for (int s = 128; s > 0; s >>= 1) {
        if (tid < s) red[tid] += red[tid + s];
        __syncthreads();
    }
    const float svv = red[0];

    const int n = blockIdx.x * 256 + tid;           // global position
    const float4* row = (const float4*)(x + (size_t)n * CC);
    float acc = 0.f;
#pragma unroll 8
    for (int j = 0; j < CC / 4; ++j) {
        float4 v = row[j];
        acc += v.x * wv_s[4*j+0] + v.y * wv_s[4*j+1]
             + v.z * wv_s[4*j+2] + v.w * wv_s[4*j+3];
    }
    const int spat = n & (HWN - 1);
    logits[n] = acc + pe[spat] * svv;
}

// ---------------------------------------------------------------------------
// K2: per-batch softmax stats: m (max), S (sum of exp), q = sum(exp*pe)/S
// stats layout: [m0..m3, S0..S3, q0..q3]
// ---------------------------------------------------------------------------
__global__ void k_softmax_stats(const float* __restrict__ logits,
                                const float* __restrict__ pe,
                                float* __restrict__ stats) {
    __shared__ float red[1024];
    const int b = blockIdx.x;
    const int tid = threadIdx.x;
    const float* lb = logits + (size_t)b * HWN;

    float lm = -3.0e38f;
    for (int i = tid; i < HWN; i += 1024) lm = fmaxf(lm, lb[i]);
    red[tid] = lm; __syncthreads();
    for (int s = 512; s > 0; s >>= 1) {
        if (tid < s) red[tid] = fmaxf(red[tid], red[tid + s]);
        __syncthreads();
    }
    const float m = red[0];
    __syncthreads();

    float ls = 0.f, lq = 0.f;
    for (int i = tid; i < HWN; i += 1024) {
        float e = expf(lb[i] - m);
        ls += e;
        lq += e * pe[i];
    }
    red[tid] = ls; __syncthreads();
    for (int s = 512; s > 0; s >>= 1) {
        if (tid < s) red[tid] += red[tid + s];
        __syncthreads();
    }
    const float S = red[0];
    __syncthreads();
    red[tid] = lq; __syncthreads();
    for (int s = 512; s > 0; s >>= 1) {
        if (tid < s) red[tid] += red[tid + s];
        __syncthreads();
    }
    if (tid == 0) {
        stats[b]     = m;
        stats[4 + b] = S;
        stats[8 + b] = red[0] / S;   // q = sum(attn * pe)
    }
}

// ---------------------------------------------------------------------------
// K3: partial weighted pooling. Block = (chunk, batch). 256 positions/chunk.
// part[(b*CHUNKS + chunk)*CC + c] = sum_p attn[p] * x[p, c]
// ---------------------------------------------------------------------------
__global__ void k_pool_partial(const float* __restrict__ x,
                               const float* __restrict__ logits,
                               const float* __restrict__ stats,
                               float* __restrict__ part) {
    __shared__ float wsh[256];
    const int chunk = blockIdx.x;
    const int b     = blockIdx.y;
    const int tid   = threadIdx.x;
    const int p0    = chunk * 256;

    const float m = stats[b];
    const float S = stats[4 + b];
    wsh[tid] = expf(logits[(size_t)b * HWN + p0 + tid] - m) / S;
    __syncthreads();

    const float* xb = x + ((size_t)b * HWN + p0) * CC;
    float acc = 0.f;
#pragma unroll 4
    for (int p = 0; p < 256; ++p)
        acc += wsh[p] * xb[(size_t)p * CC + tid];

    part[((size_t)b * CHUNKS + chunk) * CC + tid] = acc;
}

// ---------------------------------------------------------------------------
// K3b: deterministic reduce of partials + pe correction.
// pooled[b*CC + c] = q[b] + sum_chunk part[b][chunk][c]
// ---------------------------------------------------------------------------
__global__ void k_pool_reduce(const float* __restrict__ part,
                              const float* __restrict__ stats,
                              float* __restrict__ pooled) {
    const int b = blockIdx.x;
    const int c = threadIdx.x;
    float acc = stats[8 + b];
    for (int k = 0; k < CHUNKS; ++k)
        acc += part[((size_t)b * CHUNKS + k) * CC + c];
    pooled[b * CC + c] = acc;
}

// ---------------------------------------------------------------------------
// K4: excitation path. Single block, 512 threads = 16 waves.
//   context = pooled_padded(16xC) @ Wk(CxC)   via v_wmma_f32_16x16x4_f32
//   h = context @ W1 ; LayerNorm(eps=1e-3) ; clip 0..6 ; channel_add = h @ W2
// ---------------------------------------------------------------------------
__global__ void k_excite(const float* __restrict__ pooled,
                         const float* __restrict__ Wk,
                         const float* __restrict__ W1,
                         const float* __restrict__ gamma,
                         const float* __restrict__ beta,
                         const float* __restrict__ W2,
                         float* __restrict__ channel_add) {
    __shared__ float Alds[16 * CC];     // padded A, rows 4..15 zero
    __shared__ float ctx[BB * CC];
    __shared__ float hbuf[BB * MID];
    __shared__ float lnst[2 * BB];      // mu, rstd per batch

    const int tid = threadIdx.x;

    for (int i = tid; i < 16 * CC; i += 512) {
        int m = i / CC, k = i % CC;
        Alds[i] = (m < BB) ? pooled[m * CC + k] : 0.f;
    }
    __syncthreads();

    const int lane = tid & 31;
    const int wave = tid >> 5;          // 0..15 -> N tile
    const int n0   = wave * 16;
    const int hi   = lane >> 4;         // lane half selects K pair {0,1}/{2,3}
    const int nl   = lane & 15;         // row (A) / col (B) within tile

    v8f acc = {0.f, 0.f, 0.f, 0.f, 0.f, 0.f, 0.f, 0.f};
#pragma unroll 4
    for (int kk = 0; kk < CC / 4; ++kk) {
        const int kb = kk * 4;
        v2f a, bf;
        // A 16x4: VGPR0 = K=kb+2*hi, VGPR1 = K=kb+1+2*hi, M = nl
        a.x  = Alds[nl * CC + kb + 2 * hi];
        a.y  = Alds[nl * CC + kb + 1 + 2 * hi];
        // B 4x16: same K split; N = n0+nl
        bf.x = Wk[(size_t)(kb + 2 * hi) * CC + n0 + nl];
        bf.y = Wk[(size_t)(kb + 1 + 2 * hi) * CC + n0 + nl];
        acc = __builtin_amdgcn_wmma_f32_16x16x4_f32(
            false, a, false, bf, (short)0, acc, false, false);
    }
    // C/D layout: VGPR r = row M=r (lanes 0-15). Only rows 0..3 are real.
    if (hi == 0) {
        ctx[0 * CC + n0 + nl] = acc[0];
        ctx[1 * CC + n0 + nl] = acc[1];
        ctx[2 * CC + n0 + nl] = acc[2];
        ctx[3 * CC + n0 + nl] = acc[3];
    }
    __syncthreads();

    // h = ctx @ W1  (4x256 @ 256x64)
    if (tid < BB * MID) {
        const int b = tid / MID, mid = tid % MID;
        float s = 0.f;
        for (int c = 0; c < CC; ++c) s += ctx[b * CC + c] * W1[c * MID + mid];
        hbuf[tid] = s;
    }
    __syncthreads();

    // LayerNorm stats over MID=64 per batch
    if (tid < BB) {
        float mu = 0.f;
        for (int j = 0; j < MID; ++j) mu += hbuf[tid * MID + j];
        mu *= (1.f / MID);
        float var = 0.f;
        for (int j = 0; j < MID; ++j) {
            float d = hbuf[tid * MID + j] - mu;
            var += d * d;
        }
        var *= (1.f / MID);
        lnst[tid * 2 + 0] = mu;
        lnst[tid * 2 + 1] = rsqrtf(var + 1e-3f);
    }
    __syncthreads();

    if (tid < BB * MID) {
        const int b = tid / MID, mid = tid % MID;
        float v = (hbuf[tid] - lnst[b * 2]) * lnst[b * 2 + 1] * gamma[mid] + beta[mid];
        hbuf[tid] = fminf(fmaxf(v, 0.f), 6.f);
    }
    __syncthreads();

    // channel_add = h @ W2  (4x64 @ 64x256)
    for (int o = tid; o < BB * CC; o += 512) {
        const int b = o / CC, c = o % CC;
        float s = 0.f;
        for (int mid = 0; mid < MID; ++mid)
            s += hbuf[b * MID + mid] * W2[mid * CC + c];
        channel_add[o] = s;
    }
}

// ---------------------------------------------------------------------------
// K5: out = x + pe + channel_add  (float4 streaming)
// ---------------------------------------------------------------------------
__global__ void k_output(const float* __restrict__ x,
                         const float* __restrict__ pe,
                         const float* __restrict__ channel_add,
                         float* __restrict__ out) {
    const size_t i4 = (size_t)blockIdx.x * 256 + threadIdx.x;  // float4 index
    const int c4    = (int)(i4 & (CC / 4 - 1));                // 0..63
    const size_t sp = i4 >> 6;
    const int spat  = (int)(sp & (HWN - 1));
    const int b     = (int)(sp >> 16);

    const float p = pe[spat];
    float4 xv = ((const float4*)x)[i4];
    float4 ca = ((const float4*)channel_add)[b * (CC / 4) + c4];
    float4 r;
    r.x = xv.x + p + ca.x;
    r.y = xv.y + p + ca.y;
    r.z = xv.z + p + ca.z;
    r.w = xv.w + p + ca.w;
    ((float4*)out)[i4] = r;
}

// ---------------------------------------------------------------------------
extern "C" void kernel_launch(void* const* d_in, const int* in_sizes, int n_in,
                              void* d_out, int out_size, void* d_ws, size_t ws_size,
                              hipStream_t stream) {
    const float* x     = (const float*)d_in[0];
    const float* pe    = (const float*)d_in[1];
    const float* Wk    = (const float*)d_in[2];
    const float* Wv    = (const float*)d_in[3];
    const float* W1    = (const float*)d_in[4];
    const float* gamma = (const float*)d_in[5];
    const float* beta  = (const float*)d_in[6];
    const float* W2    = (const float*)d_in[7];
    float* out = (float*)d_out;

    // workspace layout (floats)
    float* ws      = (float*)d_ws;
    float* logits  = ws;                        // 262144
    float* part    = logits + NPOS;             // 262144
    float* stats   = part + (size_t)BB*CHUNKS*CC; // 12
    float* pooled  = stats + 16;                // 1024
    float* chadd   = pooled + BB * CC;          // 1024

    k_logits<<<NPOS / 256, 256, 0, stream>>>(x, pe, Wv, logits);
    k_softmax_stats<<<BB, 1024, 0, stream>>>(logits, pe, stats);
    k_pool_partial<<<dim3(CHUNKS, BB), 256, 0, stream>>>(x, logits, stats, part);
    k_pool_reduce<<<BB, CC, 0, stream>>>(part, stats, pooled);
    k_excite<<<1, 512, 0, stream>>>(pooled, Wk, W1, gamma, beta, W2, chadd);
    k_output<<<(size_t)BB * HWN * CC / 4 / 256, 256, 0, stream>>>(x, pe, chadd, out);
}